// FixedGraphSAGE_28802050687006
// MI455X (gfx1250) — compile-verified
//
#include <hip/hip_runtime.h>

typedef float v2f __attribute__((ext_vector_type(2)));
typedef float v8f __attribute__((ext_vector_type(8)));

#define NNODES 50000
#define NEDGES 800000
#define HID    128
#define LDSPAD 4
#define EPS_BN   1e-5f
#define EPS_NORM 1e-12f

// ---------------------------------------------------------------------------
// K-loop over one input operand: A tile (16 x K) staged in LDS, B = W (K x 128)
// read from global (weights are tiny -> L2/WGP$ resident).
// WMMA f32 16x16x4 fragment mapping (wave32):
//   A: lane m=lane&15 holds row m; kh=(lane>>4)*2 selects K pair {kh,kh+1}
//   B: same striping over K, n = lane&15 selects column
//   C/D: VGPR v -> row v + 8*(lane>>4), col = lane&15
// ---------------------------------------------------------------------------
__device__ __forceinline__ v8f wmma_k_loop(const float* __restrict__ ldsA, int ldsStride,
                                           int K, const float* __restrict__ W,
                                           int colbase, v8f c) {
  const int lane = threadIdx.x & 31;
  const int m    = lane & 15;
  const int kh   = (lane >> 4) << 1;           // 0 or 2
  const float* arow = ldsA + m * ldsStride + kh;
  const float* wcol = W + colbase + m;         // m doubles as n for B/C
#pragma unroll 8
  for (int k0 = 0; k0 < K; k0 += 4) {
    v2f a, b;
    a.x = arow[k0 + 0];
    a.y = arow[k0 + 1];
    b.x = wcol[(k0 + kh    ) * HID];
    b.y = wcol[(k0 + kh + 1) * HID];
    c = __builtin_amdgcn_wmma_f32_16x16x4_f32(false, a, false, b, (short)0, c,
                                              false, false);
  }
  return c;
}

// ---------------------------------------------------------------------------
// Fused GEMM: D[16 x 128 per block] = sum_i A_i @ W_i (+bias0+bias1)
//   A0 may be row-scaled (mean aggregation 1/deg), optional leaky ReLU,
//   optional BatchNorm column-stat accumulation (sum, sumsq) via atomics.
// ---------------------------------------------------------------------------
__global__ __launch_bounds__(256) void gemm_fused_kernel(
    const float* __restrict__ A0, const float* __restrict__ W0, int K0,
    const float* __restrict__ rowscale0,
    const float* __restrict__ A1, const float* __restrict__ W1, int K1,
    const float* __restrict__ A2, const float* __restrict__ W2, int K2,
    const float* __restrict__ bias0, const float* __restrict__ bias1,
    float* __restrict__ D, float* __restrict__ colsum, float* __restrict__ colsq,
    int applyLeaky)
{
  __shared__ float ldsA[16 * (256 + LDSPAD)];
  const int r0      = blockIdx.x * 16;
  const int tid     = threadIdx.x;
  const int lane    = tid & 31;
  const int wv      = tid >> 5;           // 0..7
  const int colbase = wv * 16;

  v8f c = {0.f, 0.f, 0.f, 0.f, 0.f, 0.f, 0.f, 0.f};

  // ---- input 0 (optional row scaling) ----
  {
    const int K = K0, S = K0 + LDSPAD;
    for (int t = tid; t < 16 * K; t += 256) {
      int row = t / K, col = t - row * K;
      float v = A0[(size_t)(r0 + row) * K + col];
      if (rowscale0) v *= rowscale0[r0 + row];
      ldsA[row * S + col] = v;
    }
    __syncthreads();
    c = wmma_k_loop(ldsA, S, K, W0, colbase, c);
    __syncthreads();
  }
  // ---- input 1 ----
  if (A1) {
    const int K = K1, S = K1 + LDSPAD;
    for (int t = tid; t < 16 * K; t += 256) {
      int row = t / K, col = t - row * K;
      ldsA[row * S + col] = A1[(size_t)(r0 + row) * K + col];
    }
    __syncthreads();
    c = wmma_k_loop(ldsA, S, K, W1, colbase, c);
    __syncthreads();
  }
  // ---- input 2 ----
  if (A2) {
    const int K = K2, S = K2 + LDSPAD;
    for (int t = tid; t < 16 * K; t += 256) {
      int row = t / K, col = t - row * K;
      ldsA[row * S + col] = A2[(size_t)(r0 + row) * K + col];
    }
    __syncthreads();
    c = wmma_k_loop(ldsA, S, K, W2, colbase, c);
    __syncthreads();
  }

  // ---- epilogue: bias, leaky, store, BN column stats ----
  const int n = colbase + (lane & 15);
  float bias = 0.f;
  if (bias0) bias += bias0[n];
  if (bias1) bias += bias1[n];

  const int rowoff = (lane >> 4) * 8;
  float psum = 0.f, psq = 0.f;
#pragma unroll
  for (int v = 0; v < 8; ++v) {
    float x = c[v] + bias;
    if (applyLeaky) x = (x >= 0.f) ? x : 0.2f * x;
    D[(size_t)(r0 + rowoff + v) * HID + n] = x;
    psum += x;
    psq  += x * x;
  }
  if (colsum) {
    psum += __shfl_xor(psum, 16, 32);   // combine rows 0-7 with rows 8-15
    psq  += __shfl_xor(psq, 16, 32);
    if (lane < 16) {
      atomicAdd(&colsum[n], psum);
      atomicAdd(&colsq[n],  psq);
    }
  }
}

// ---------------------------------------------------------------------------
__global__ void zero_kernel(float* __restrict__ p, size_t n) {
  size_t i = (size_t)blockIdx.x * 256 + threadIdx.x;
  if (i < n) p[i] = 0.f;
}

__global__ void count_kernel(const int* __restrict__ dst, float* __restrict__ cnt, int nE) {
  int i = blockIdx.x * 256 + threadIdx.x;
  if (i < nE) atomicAdd(&cnt[dst[i]], 1.0f);
}

__global__ void invcnt_kernel(float* __restrict__ cnt, int n) {
  int i = blockIdx.x * 256 + threadIdx.x;
  if (i < n) cnt[i] = 1.0f / fmaxf(cnt[i], 1.0f);
}

// one wave per edge; each lane moves one float4 (128 floats / edge)
__global__ __launch_bounds__(256) void scatter_kernel(
    const float* __restrict__ feat, const int* __restrict__ src,
    const int* __restrict__ dst, float* __restrict__ agg, int nE)
{
  int e    = (blockIdx.x * 256 + threadIdx.x) >> 5;
  int lane = threadIdx.x & 31;
  if (e >= nE) return;
  int s = src[e], d = dst[e];
  const float4 v = ((const float4*)(feat + (size_t)s * HID))[lane];
  float* out = agg + (size_t)d * HID + lane * 4;
  atomicAdd(out + 0, v.x);
  atomicAdd(out + 1, v.y);
  atomicAdd(out + 2, v.z);
  atomicAdd(out + 3, v.w);
}

__global__ void bn_finalize_kernel(const float* __restrict__ colsum,
                                   const float* __restrict__ colsq,
                                   const float* __restrict__ g,
                                   const float* __restrict__ be,
                                   float* __restrict__ scale,
                                   float* __restrict__ shift, float invN) {
  int i = threadIdx.x;                       // 128 threads
  float mean = colsum[i] * invN;
  float var  = colsq[i] * invN - mean * mean;
  float sc   = g[i] * rsqrtf(var + EPS_BN);
  scale[i] = sc;
  shift[i] = be[i] - mean * sc;
}

__global__ void bn_apply_kernel(float* __restrict__ h, const float* __restrict__ scale,
                                const float* __restrict__ shift, size_t n) {
  size_t i = (size_t)blockIdx.x * 256 + threadIdx.x;
  if (i < n) {
    int col = (int)(i & (HID - 1));
    h[i] = h[i] * scale[col] + shift[col];
  }
}

// BN2 + row L2 normalize, in place on d_out; one 128-thread block per row
__global__ __launch_bounds__(128) void final_norm_kernel(
    float* __restrict__ out, const float* __restrict__ scale,
    const float* __restrict__ shift)
{
  __shared__ float red[4];
  const int row = blockIdx.x;
  const int c   = threadIdx.x;
  float v = out[(size_t)row * HID + c] * scale[c] + shift[c];
  float s = v * v;
#pragma unroll
  for (int off = 16; off >= 1; off >>= 1) s += __shfl_xor(s, off, 32);
  int lane = c & 31, wv = c >> 5;
  if (lane == 0) red[wv] = s;
  __syncthreads();
  float nrm = sqrtf(red[0] + red[1] + red[2] + red[3]);
  out[(size_t)row * HID + c] = v / fmaxf(nrm, EPS_NORM);
}

// ---------------------------------------------------------------------------
extern "C" void kernel_launch(void* const* d_in, const int* in_sizes, int n_in,
                              void* d_out, int out_size, void* d_ws, size_t ws_size,
                              hipStream_t stream) {
  const float* x     = (const float*)d_in[0];
  const int*   ei    = (const int*)  d_in[1];
  const float* W_in  = (const float*)d_in[2];
  const float* b_in  = (const float*)d_in[3];
  const float* g1    = (const float*)d_in[4];
  const float* be1   = (const float*)d_in[5];
  const float* Wl1   = (const float*)d_in[6];
  const float* bl1   = (const float*)d_in[7];
  const float* Wr1   = (const float*)d_in[8];
  const float* Wl2   = (const float*)d_in[9];
  const float* bl2   = (const float*)d_in[10];
  const float* Wr2   = (const float*)d_in[11];
  const float* Wskip = (const float*)d_in[12];
  const float* bskip = (const float*)d_in[13];
  const float* g2    = (const float*)d_in[14];
  const float* be2   = (const float*)d_in[15];
  float* out = (float*)d_out;
  float* ws  = (float*)d_ws;

  const int* srcIdx = ei;
  const int* dstIdx = ei + NEDGES;

  const size_t NF = (size_t)NNODES * HID;
  float* h     = ws;
  float* h1    = ws + NF;
  float* agg   = ws + 2 * NF;
  float* cnt   = ws + 3 * NF;              // later holds 1/max(deg,1)
  float* stats = cnt + NNODES;
  float *sum1 = stats,       *sq1 = stats + 128, *sc1 = stats + 256, *sh1 = stats + 384;
  float *sum2 = stats + 512, *sq2 = stats + 640, *sc2 = stats + 768, *sh2 = stats + 896;

  // zero agg + cnt + stats (contiguous)
  size_t zc = NF + NNODES + 1024;
  zero_kernel<<<(unsigned)((zc + 255) / 256), 256, 0, stream>>>(agg, zc);

  // h_pre = leaky(x @ W_in + b_in), accumulate BN1 stats
  gemm_fused_kernel<<<NNODES / 16, 256, 0, stream>>>(
      x, W_in, 256, nullptr,
      nullptr, nullptr, 0, nullptr, nullptr, 0,
      b_in, nullptr, h, sum1, sq1, 1);
  bn_finalize_kernel<<<1, 128, 0, stream>>>(sum1, sq1, g1, be1, sc1, sh1, 1.0f / NNODES);
  bn_apply_kernel<<<(unsigned)(NF / 256), 256, 0, stream>>>(h, sc1, sh1, NF);

  // degrees -> 1/max(deg,1)
  count_kernel<<<(NEDGES + 255) / 256, 256, 0, stream>>>(dstIdx, cnt, NEDGES);
  invcnt_kernel<<<(NNODES + 255) / 256, 256, 0, stream>>>(cnt, NNODES);

  // conv1: agg = scatter-sum(h[src] -> dst); h1 = leaky(agg/deg @ Wl1 + bl1 + h @ Wr1)
  scatter_kernel<<<(NEDGES * 32) / 256, 256, 0, stream>>>(h, srcIdx, dstIdx, agg, NEDGES);
  gemm_fused_kernel<<<NNODES / 16, 256, 0, stream>>>(
      agg, Wl1, 128, cnt,
      h, Wr1, 128, nullptr, nullptr, 0,
      bl1, nullptr, h1, nullptr, nullptr, 1);

  // conv2 + skip: out_pre = agg2/deg @ Wl2 + h1 @ Wr2 + h @ Wskip + bl2 + bskip
  zero_kernel<<<(unsigned)((NF + 255) / 256), 256, 0, stream>>>(agg, NF);
  scatter_kernel<<<(NEDGES * 32) / 256, 256, 0, stream>>>(h1, srcIdx, dstIdx, agg, NEDGES);
  gemm_fused_kernel<<<NNODES / 16, 256, 0, stream>>>(
      agg, Wl2, 128, cnt,
      h1, Wr2, 128, h, Wskip, 128,
      bl2, bskip, out, sum2, sq2, 0);

  // BN2 + row normalize (in place on d_out)
  bn_finalize_kernel<<<1, 128, 0, stream>>>(sum2, sq2, g2, be2, sc2, sh2, 1.0f / NNODES);
  final_norm_kernel<<<NNODES, 128, 0, stream>>>(out, sc2, sh2);
}